// My_GAT_22857815950049
// MI455X (gfx1250) — compile-verified
//
#include <hip/hip_runtime.h>
#include <math.h>

#define N_NODES 100000
#define N_EDGES 1600000
#define DIN 24
#define DH 64
#define DOUT 12
#define NEG_SLOPE 0.01f

typedef __attribute__((ext_vector_type(2))) float v2f;
typedef __attribute__((ext_vector_type(8))) float v8f;

// ---------------- float atomic max via ordered integer bits ----------------
__device__ __forceinline__ void atomicMaxF32(float* addr, float val) {
    if (val >= 0.0f) {
        atomicMax((int*)addr, __float_as_int(val));
    } else {
        atomicMin((unsigned int*)addr, (unsigned int)__float_as_int(val));
    }
}

// ---------------- h = feats @ emb_h_W.T + emb_h_b (N x 24 -> N x 64) -------
__global__ void k_embed_h(const float* __restrict__ feats,
                          const float* __restrict__ W,
                          const float* __restrict__ b,
                          float* __restrict__ h) {
    int i = blockIdx.x * blockDim.x + threadIdx.x;
    if (i >= N_NODES * DH) return;
    int n = i >> 6, o = i & 63;
    const float* fr = feats + n * DIN;
    const float* wr = W + o * DIN;
    float acc = b[o];
#pragma unroll
    for (int k = 0; k < DIN; ++k) acc = fmaf(fr[k], wr[k], acc);
    h[i] = acc;
}

// ------------- dual GEMM: h_s = h@Wself.T, z = h@Wfunc.T (WMMA f32) --------
// Block = 256 threads = 8 waves. Waves 0-3 -> Wself tiles, 4-7 -> Wfunc tiles.
// Each wave computes one 16x16 output tile via 16x V_WMMA_F32_16X16X4_F32.
// A layout (16x4 f32): lanes 0-15 hold K={0,1} in v[0..1], lanes 16-31 K={2,3}.
// B layout (4x16): lane holds B[k][n], n = lane&15, k split like A.
// C/D layout: VGPR r = row (r + 8*(lane>>4)), col = lane&15.
__global__ void __launch_bounds__(256)
k_gemm_dual(const float* __restrict__ h,
            const float* __restrict__ Wself,
            const float* __restrict__ Wfunc,
            float* __restrict__ hs,
            float* __restrict__ z) {
    const int node_base = blockIdx.x * 16;
    const int wave = threadIdx.x >> 5;
    const int lane = threadIdx.x & 31;
    const int half = lane >> 4;   // 0 or 1
    const int l16  = lane & 15;

    const float* W  = (wave < 4) ? Wself : Wfunc;
    float* out      = (wave < 4) ? hs : z;
    const int o_base = (wave & 3) * 16;

    const float* arow = h + (node_base + l16) * DH; // A row for this lane
    const float* brow = W + (o_base + l16) * DH;    // B[k][n] = W[n][k]

    v8f acc = {};
#pragma unroll
    for (int k0 = 0; k0 < 16; ++k0) {
        const int kk = k0 * 4 + half * 2;          // even -> 8B aligned
        v2f a = *(const v2f*)(arow + kk);
        v2f b = *(const v2f*)(brow + kk);
        acc = __builtin_amdgcn_wmma_f32_16x16x4_f32(
            /*neg_a=*/false, a, /*neg_b=*/false, b,
            /*c_mod=*/(short)0, acc, /*reuse_a=*/false, /*reuse_b=*/false);
    }
#pragma unroll
    for (int r = 0; r < 8; ++r) {
        int m = node_base + r + half * 8;
        out[m * DH + o_base + l16] = acc[r];
    }
}

// ---------------- s_src[n] = z[n]·a_src, s_dst[n] = z[n]·a_dst -------------
__global__ void k_node_scores(const float* __restrict__ z,
                              const float* __restrict__ attn,
                              float* __restrict__ ssrc,
                              float* __restrict__ sdst) {
    int n = blockIdx.x * blockDim.x + threadIdx.x;
    if (n >= N_NODES) return;
    const float* zr = z + n * DH;
    float a = 0.0f, b = 0.0f;
#pragma unroll
    for (int k = 0; k < DH; ++k) {
        float v = zr[k];
        a = fmaf(v, attn[k], a);
        b = fmaf(v, attn[DH + k], b);
    }
    ssrc[n] = a;
    sdst[n] = b;
}

// ------- c1 = sum_d emb_e_W[d]*a_w[d], c0 = sum_d emb_e_b[d]*a_w[d] --------
__global__ void k_edge_coefs(const float* __restrict__ eW,
                             const float* __restrict__ eb,
                             const float* __restrict__ attn,
                             float* __restrict__ coef) {
    __shared__ float s1[DH];
    __shared__ float s0[DH];
    int t = threadIdx.x;
    float aw = attn[2 * DH + t];
    s1[t] = eW[t] * aw;
    s0[t] = eb[t] * aw;
    __syncthreads();
    for (int s = DH / 2; s > 0; s >>= 1) {
        if (t < s) { s1[t] += s1[t + s]; s0[t] += s0[t + s]; }
        __syncthreads();
    }
    if (t == 0) { coef[0] = s1[0]; coef[1] = s0[0]; }
}

// ---------------- per-layer init: agg=0, m=-inf, denom=0 -------------------
__global__ void k_init_layer(float* __restrict__ agg,
                             float* __restrict__ m,
                             float* __restrict__ denom) {
    int i = blockIdx.x * blockDim.x + threadIdx.x;
    if (i < N_NODES * DH) agg[i] = 0.0f;
    if (i < N_NODES) { m[i] = -INFINITY; denom[i] = 0.0f; }
}

__global__ void k_indeg_init(float* __restrict__ indeg) {
    int i = blockIdx.x * blockDim.x + threadIdx.x;
    if (i < N_NODES) indeg[i] = 0.0f;
}

__global__ void k_indeg(const int* __restrict__ dst, float* __restrict__ indeg) {
    int e = blockIdx.x * blockDim.x + threadIdx.x;
    if (e >= N_EDGES) return;
    atomicAdd(&indeg[dst[e]], 1.0f);
}

// ------- edge scores + leaky relu + segment max (atomic float max) ---------
__global__ void k_edge_score(const int* __restrict__ src,
                             const int* __restrict__ dst,
                             const float* __restrict__ ew,
                             const float* __restrict__ ssrc,
                             const float* __restrict__ sdst,
                             const float* __restrict__ coef,
                             float* __restrict__ score,
                             float* __restrict__ m) {
    int e = blockIdx.x * blockDim.x + threadIdx.x;
    if (e >= N_EDGES) return;
    float v = ssrc[src[e]] + sdst[dst[e]] + fmaf(coef[0], ew[e], coef[1]);
    v = (v > 0.0f) ? v : NEG_SLOPE * v;
    score[e] = v;
    atomicMaxF32(&m[dst[e]], v);
}

// ---------------- ex = exp(score - m[dst]); denom += ex --------------------
__global__ void k_edge_exp(const int* __restrict__ dst,
                           const float* __restrict__ score,
                           const float* __restrict__ m,
                           float* __restrict__ ex,
                           float* __restrict__ denom) {
    int e = blockIdx.x * blockDim.x + threadIdx.x;
    if (e >= N_EDGES) return;
    int d = dst[e];
    float v = __expf(score[e] - m[d]);
    ex[e] = v;
    atomicAdd(&denom[d], v);
}

// -------- agg[dst] += (ex/denom[dst]) * z[src]  (thread per (edge,dim)) ----
__global__ void k_edge_agg(const int* __restrict__ src,
                           const int* __restrict__ dst,
                           const float* __restrict__ ex,
                           const float* __restrict__ denom,
                           const float* __restrict__ z,
                           float* __restrict__ agg) {
    int i = blockIdx.x * blockDim.x + threadIdx.x; // E*DH = 102.4M < 2^31
    if (i >= N_EDGES * DH) return;
    int e = i >> 6, d = i & 63;
    int de = dst[e];
    float alpha = ex[e] / denom[de];
    atomicAdd(&agg[de * DH + d], alpha * z[src[e] * DH + d]);
}

// ---------------- h = h + relu(in_deg>0 ? h_s+agg : h) ---------------------
__global__ void k_node_update(const float* __restrict__ hs,
                              const float* __restrict__ agg,
                              const float* __restrict__ indeg,
                              float* __restrict__ h) {
    int i = blockIdx.x * blockDim.x + threadIdx.x;
    if (i >= N_NODES * DH) return;
    int n = i >> 6;
    float hv = h[i];
    float mid = (indeg[n] > 0.0f) ? (hs[i] + agg[i]) : hv;
    h[i] = hv + fmaxf(mid, 0.0f);
}

// ---------------- y = h @ lin1_W.T + lin1_b (N x 64 -> N x 12) -------------
__global__ void k_final(const float* __restrict__ h,
                        const float* __restrict__ W,
                        const float* __restrict__ b,
                        float* __restrict__ y) {
    int i = blockIdx.x * blockDim.x + threadIdx.x;
    if (i >= N_NODES * DOUT) return;
    int n = i / DOUT, o = i % DOUT;
    const float* hr = h + n * DH;
    const float* wr = W + o * DH;
    float acc = b[o];
#pragma unroll
    for (int k = 0; k < DH; ++k) acc = fmaf(hr[k], wr[k], acc);
    y[i] = acc;
}

// ===========================================================================
extern "C" void kernel_launch(void* const* d_in, const int* in_sizes, int n_in,
                              void* d_out, int out_size, void* d_ws, size_t ws_size,
                              hipStream_t stream) {
    const float* feats   = (const float*)d_in[0];
    const float* e_w     = (const float*)d_in[1];
    // d_in[2] snorm_n, d_in[3] snorm_e unused by reference
    const int*   src     = (const int*)d_in[4];
    const int*   dst     = (const int*)d_in[5];
    const float* emb_h_W = (const float*)d_in[6];
    const float* emb_h_b = (const float*)d_in[7];
    const float* emb_e_W = (const float*)d_in[8];
    const float* emb_e_b = (const float*)d_in[9];
    const float* W_self1 = (const float*)d_in[10];
    const float* W_func1 = (const float*)d_in[11];
    const float* attn1   = (const float*)d_in[12];
    const float* W_self2 = (const float*)d_in[13];
    const float* W_func2 = (const float*)d_in[14];
    const float* attn2   = (const float*)d_in[15];
    const float* lin1_W  = (const float*)d_in[16];
    const float* lin1_b  = (const float*)d_in[17];
    float* out = (float*)d_out;

    // ---- workspace layout (floats) ----
    float* ws    = (float*)d_ws;
    float* h     = ws;  ws += (size_t)N_NODES * DH;
    float* hs    = ws;  ws += (size_t)N_NODES * DH;
    float* z     = ws;  ws += (size_t)N_NODES * DH;
    float* agg   = ws;  ws += (size_t)N_NODES * DH;
    float* ssrc  = ws;  ws += N_NODES;
    float* sdst  = ws;  ws += N_NODES;
    float* m     = ws;  ws += N_NODES;
    float* denom = ws;  ws += N_NODES;
    float* indeg = ws;  ws += N_NODES;
    float* score = ws;  ws += N_EDGES;
    float* ex    = ws;  ws += N_EDGES;
    float* coef  = ws;  ws += 2;

    const int B = 256;
    const int gNode   = (N_NODES + B - 1) / B;            // 391
    const int gNodeD  = (N_NODES * DH + B - 1) / B;       // 25000
    const int gEdge   = (N_EDGES + B - 1) / B;            // 6250
    const int gEdgeD  = (N_EDGES * DH + B - 1) / B;       // 400000
    const int gGemm   = N_NODES / 16;                     // 6250 (exact)
    const int gOut    = (N_NODES * DOUT + B - 1) / B;     // 4688

    // embedding + in-degree (layer-invariant)
    k_embed_h<<<gNodeD, B, 0, stream>>>(feats, emb_h_W, emb_h_b, h);
    k_indeg_init<<<gNode, B, 0, stream>>>(indeg);
    k_indeg<<<gEdge, B, 0, stream>>>(dst, indeg);

    const float* Ws[2]  = { W_self1, W_self2 };
    const float* Wf[2]  = { W_func1, W_func2 };
    const float* at[2]  = { attn1, attn2 };

    for (int layer = 0; layer < 2; ++layer) {
        k_gemm_dual<<<gGemm, B, 0, stream>>>(h, Ws[layer], Wf[layer], hs, z);
        k_node_scores<<<gNode, B, 0, stream>>>(z, at[layer], ssrc, sdst);
        k_edge_coefs<<<1, DH, 0, stream>>>(emb_e_W, emb_e_b, at[layer], coef);
        k_init_layer<<<gNodeD, B, 0, stream>>>(agg, m, denom);
        k_edge_score<<<gEdge, B, 0, stream>>>(src, dst, e_w, ssrc, sdst, coef, score, m);
        k_edge_exp<<<gEdge, B, 0, stream>>>(dst, score, m, ex, denom);
        k_edge_agg<<<gEdgeD, B, 0, stream>>>(src, dst, ex, denom, z, agg);
        k_node_update<<<gNodeD, B, 0, stream>>>(hs, agg, indeg, h);
    }

    k_final<<<gOut, B, 0, stream>>>(h, lin1_W, lin1_b, out);
}